// augmentaion_2113123910299
// MI455X (gfx1250) — compile-verified
//
#include <hip/hip_runtime.h>
#include <math.h>

// CDNA5 wave32 WMMA types
typedef __attribute__((ext_vector_type(2))) float v2f;
typedef __attribute__((ext_vector_type(8))) float v8f;

#define BATCHES           512
#define SEQ               512
#define DIMS              150
#define FLOATS_PER_BATCH  (SEQ * DIMS)            // 76800 floats per batch
#define POINTS_PER_BATCH  (SEQ * (DIMS / 3))      // 25600 points per batch
#define CHUNKS_PER_BATCH  (POINTS_PER_BATCH / 16) // 1600 chunks of 16 points
#define BLOCKS_PER_BATCH  4
#define CHUNKS_PER_BLOCK  (CHUNKS_PER_BATCH / BLOCKS_PER_BATCH) // 400
#define WAVES_PER_BLOCK   8                       // 256 threads / wave32
#define WAVE_STRIDE_F     (48 * WAVES_PER_BLOCK)  // floats advanced per iter

__global__ __launch_bounds__(256)
void rot_wmma_kernel(const float* __restrict__ x,
                     const float* __restrict__ theta,
                     float* __restrict__ out)
{
    const int block = blockIdx.x;
    const int b     = block / BLOCKS_PER_BATCH;   // batch index
    const int q     = block % BLOCKS_PER_BATCH;   // quarter of the batch
    const int lane  = threadIdx.x & 31;
    const int wave  = threadIdx.x >> 5;

    // ---- per-batch rotation matrix R = Rz * Ry * Rx ------------------------
    // Reference indexing: Rx uses th[:,1], Ry uses th[:,0], Rz uses th[:,2].
    // |theta| <= 0.3, so hardware v_sin/v_cos (__sinf/__cosf) is accurate here
    // and avoids the ~1k-instruction libm Payne-Hanek path.
    const float ty = theta[b * 3 + 0];
    const float tx = theta[b * 3 + 1];
    const float tz = theta[b * 3 + 2];
    const float cx = __cosf(tx), sx = __sinf(tx);
    const float cy = __cosf(ty), sy = __sinf(ty);
    const float cz = __cosf(tz), sz = __sinf(tz);

    const float r00 = cz * cy;
    const float r01 = cz * sy * sx - sz * cx;
    const float r02 = cz * sy * cx + sz * sx;
    const float r10 = sz * cy;
    const float r11 = sz * sy * sx + cz * cx;
    const float r12 = sz * sy * cx - cz * sx;
    const float r20 = -sy;
    const float r21 = cy * sx;
    const float r22 = cy * cx;

    // ---- A matrix: 16x4, rows 0..2 = R, everything else 0 ------------------
    // 32-bit A layout (ISA 7.12.2): lane m<16 holds {A[m][0], A[m][1]},
    // lane m+16 holds {A[m][2], A[m][3]}.
    v2f a; a.x = 0.0f; a.y = 0.0f;
    if      (lane == 0)  { a.x = r00; a.y = r01; }
    else if (lane == 1)  { a.x = r10; a.y = r11; }
    else if (lane == 2)  { a.x = r20; a.y = r21; }
    else if (lane == 16) { a.x = r02; }           // A[0][2], A[0][3]=0
    else if (lane == 17) { a.x = r12; }
    else if (lane == 18) { a.x = r22; }

    const int  n16 = lane & 15;
    const bool lo  = lane < 16;

    // Per-lane pointers, bumped by a constant stride each iteration.
    // Lo lane n reads {x,y} at +3n; hi lane n reads {y,z} at +3n+1 (shifted
    // by -1 from z so BOTH loads are unconditional and always in-bounds).
    const int first_chunk = q * CHUNKS_PER_BLOCK + wave;
    const size_t base     = (size_t)b * FLOATS_PER_BATCH + (size_t)first_chunk * 48;
    const float* px = x   + base + 3 * n16 + (lo ? 0 : 1);
    float*       po = out + base + 3 * n16;       // store addr (lo lanes)

    const int iters = CHUNKS_PER_BLOCK / WAVES_PER_BLOCK;   // 50

    for (int it = 0; it < iters; ++it) {
        // Prefetch this wave's next chunk (speculative; OOB is dropped).
        __builtin_prefetch(px + WAVE_STRIDE_F, 0, 2);

        // Unconditional pair load; loop-invariant-mask selects afterwards.
        const float v0 = px[0];
        const float v1 = px[1];

        // ---- B matrix: 4x16, column n = point n, K=3 row = 0 ---------------
        // 32-bit B layout: lane n<16 holds {B[0][n], B[1][n]} = {x, y},
        // lane n+16 holds {B[2][n], B[3][n]} = {z, 0}.
        v2f bv;
        bv.x = lo ? v0 : v1;    // x (lo) / z (hi)
        bv.y = lo ? v1 : 0.0f;  // y (lo) / K=3 zero (hi)

        // D[m][n] = sum_k R[m][k] * p_n[k]  (rows 0..2 valid)
        v8f c = {};
        c = __builtin_amdgcn_wmma_f32_16x16x4_f32(
                /*neg_a=*/false, a,
                /*neg_b=*/false, bv,
                /*c_mod=*/(short)0, c,
                /*reuse_a=*/false, /*reuse_b=*/false);

        // C/D layout: lane n<16, vgpr r -> D[r][n]. Lane n stores point n.
        if (lo) {
            po[0] = c[0];
            po[1] = c[1];
            po[2] = c[2];
        }

        px += WAVE_STRIDE_F;
        po += WAVE_STRIDE_F;
    }
}

extern "C" void kernel_launch(void* const* d_in, const int* in_sizes, int n_in,
                              void* d_out, int out_size, void* d_ws, size_t ws_size,
                              hipStream_t stream) {
    const float* x     = (const float*)d_in[0];
    const float* theta = (const float*)d_in[1];
    float*       out   = (float*)d_out;

    dim3 grid(BATCHES * BLOCKS_PER_BATCH);
    dim3 block(256);
    hipLaunchKernelGGL(rot_wmma_kernel, grid, block, 0, stream, x, theta, out);
}